// Lifted_struct_loss_87935160418857
// MI455X (gfx1250) — compile-verified
//
#include <hip/hip_runtime.h>
#include <math.h>

// ---------------------------------------------------------------------------
// Lifted structured loss, fused for gfx1250 (MI455X), wave32.
//   X_embed: [4096, 512] fp32   target: [4096] int
//   out: scalar fp32
//
// ws layout (floats):
//   [0      .. 4095 ]  sqn[i]     = ||X_i||^2
//   [4096   .. 8191 ]  l_n[i]     = sum_j diff-class exp(margin - d_ij)
//   [8192]             loss_sum
//   [8193]             pair_count
//   [8194]             l2_sum (sum of ||X_i||)
// ---------------------------------------------------------------------------

#define N_EMB 4096
#define D_EMB 512
#define MARGIN 0.05f
#define L2_REG 0.02f
#define NPOS 3

#define TILE 128
#define KT 16
#define NKT (D_EMB / KT)   // 32 K-stages
#define LSTR 20   // padded LDS row stride (floats): 80B rows, 16B-aligned, bank-safe

typedef __attribute__((ext_vector_type(2))) float v2f;
typedef __attribute__((ext_vector_type(8))) float v8f;
typedef __attribute__((ext_vector_type(4))) int v4i;

typedef __attribute__((address_space(1))) v4i* as1_v4i;
typedef __attribute__((address_space(3))) v4i* as3_v4i;

// --- CDNA5 async global->LDS (ASYNCcnt path), builtin if present, asm else --
__device__ __forceinline__ void async_ld_b128(const float* gaddr, float* laddr) {
#if __has_builtin(__builtin_amdgcn_global_load_async_to_lds_b128)
    __builtin_amdgcn_global_load_async_to_lds_b128(
        (as1_v4i)gaddr, (as3_v4i)laddr, 0, 0);
#else
    // generic-pointer low 32 bits == LDS byte offset (aperture layout)
    unsigned lds_off = (unsigned)(uintptr_t)laddr;
    unsigned long long ga = (unsigned long long)(uintptr_t)gaddr;
    asm volatile("global_load_async_to_lds_b128 %0, %1, off"
                 :: "v"(lds_off), "v"(ga) : "memory");
#endif
}

template <int N>
__device__ __forceinline__ void wait_asynccnt() {
#if __has_builtin(__builtin_amdgcn_s_wait_asynccnt)
    __builtin_amdgcn_s_wait_asynccnt(N);
#else
    asm volatile("s_wait_asynccnt %0" :: "n"(N) : "memory");
#endif
}

// ---------------------------------------------------------------- init ------
__global__ void init_ws_kernel(float* __restrict__ ws) {
    int t = blockIdx.x * blockDim.x + threadIdx.x;
    if (t < N_EMB) ws[N_EMB + t] = 0.0f;     // l_n
    if (t < 3)     ws[2 * N_EMB + t] = 0.0f; // accumulators
}

// ------------------------------------------------------------ row norms -----
__global__ __launch_bounds__(256)
void row_norms_kernel(const float* __restrict__ X, float* __restrict__ ws) {
    int wave = (int)((blockIdx.x * blockDim.x + threadIdx.x) >> 5);
    int lane = threadIdx.x & 31;
    if (wave >= N_EMB) return;
    const float4* Xr = (const float4*)(X + (size_t)wave * D_EMB);
    float s = 0.0f;
#pragma unroll
    for (int t = 0; t < 4; ++t) {
        float4 v = Xr[lane + 32 * t];
        s += v.x * v.x + v.y * v.y + v.z * v.z + v.w * v.w;
    }
#pragma unroll
    for (int off = 16; off >= 1; off >>= 1) s += __shfl_xor(s, off, 32);
    if (lane == 0) {
        ws[wave] = s;
        atomicAdd(&ws[2 * N_EMB + 2], sqrtf(s));
    }
}

// ------------------------------------------------- fused Gram + exp rowsum --
// Upper-triangular 128x128 blocks of X@X^T via V_WMMA_F32_16X16X4_F32, with
// double-buffered GLOBAL_LOAD_ASYNC_TO_LDS_B128 staging (ASYNCcnt pipeline).
__global__ __launch_bounds__(256)
void gram_exp_rowsum_kernel(const float* __restrict__ X,
                            const int* __restrict__ tgt,
                            float* __restrict__ ws) {
    __shared__ __align__(16) float Alds[2][TILE * LSTR];
    __shared__ __align__(16) float Blds[2][TILE * LSTR];
    __shared__ float lnrow[TILE];
    __shared__ float lncol[TILE];

    const float* sqn = ws;
    float* l_n = ws + N_EMB;

    // map flat block id -> (bi, bj) with bi <= bj
    const int NB = N_EMB / TILE;                 // 32
    int rem = blockIdx.x;
    int bi = 0;
    while (rem >= NB - bi) { rem -= NB - bi; ++bi; }
    int bj = bi + rem;
    const int ibase = bi * TILE;
    const int jbase = bj * TILE;

    const int tid  = threadIdx.x;
    const int lane = tid & 31;
    const int wid  = tid >> 5;       // 0..7
    const int waveR = wid >> 1;      // 0..3 -> 32 output rows each
    const int waveC = wid & 1;       // 0..1 -> 64 output cols each
    const int hh   = lane >> 4;      // half of wave
    const int nloc = lane & 15;

    if (tid < TILE) { lnrow[tid] = 0.0f; lncol[tid] = 0.0f; }

    // this thread's two (row, 4-float-chunk) slots per tile
    const int row0 = tid >> 2, c40 = (tid & 3) * 4;              // lin = tid
    const int row1 = (tid + 256) >> 2, c41 = ((tid + 256) & 3) * 4;

    v8f acc[2][4];
#pragma unroll
    for (int sr = 0; sr < 2; ++sr)
#pragma unroll
        for (int sc = 0; sc < 4; ++sc)
            acc[sr][sc] = (v8f){0.f, 0.f, 0.f, 0.f, 0.f, 0.f, 0.f, 0.f};

    // prologue: stage K-stage 0 into buffer 0 (4 async b128 per thread)
    async_ld_b128(X + (size_t)(ibase + row0) * D_EMB + c40, &Alds[0][row0 * LSTR + c40]);
    async_ld_b128(X + (size_t)(jbase + row0) * D_EMB + c40, &Blds[0][row0 * LSTR + c40]);
    async_ld_b128(X + (size_t)(ibase + row1) * D_EMB + c41, &Alds[0][row1 * LSTR + c41]);
    async_ld_b128(X + (size_t)(jbase + row1) * D_EMB + c41, &Blds[0][row1 * LSTR + c41]);

    for (int kt = 0; kt < NKT; ++kt) {
        const int cur = kt & 1;
        if (kt + 1 < NKT) {
            // issue next stage into the other buffer, then drain current stage
            const int nxt = cur ^ 1;
            const int ko = (kt + 1) * KT;
            async_ld_b128(X + (size_t)(ibase + row0) * D_EMB + ko + c40, &Alds[nxt][row0 * LSTR + c40]);
            async_ld_b128(X + (size_t)(jbase + row0) * D_EMB + ko + c40, &Blds[nxt][row0 * LSTR + c40]);
            async_ld_b128(X + (size_t)(ibase + row1) * D_EMB + ko + c41, &Alds[nxt][row1 * LSTR + c41]);
            async_ld_b128(X + (size_t)(jbase + row1) * D_EMB + ko + c41, &Blds[nxt][row1 * LSTR + c41]);
            wait_asynccnt<4>();   // stage kt landed (4 newest still in flight)
        } else {
            wait_asynccnt<0>();
        }
        __syncthreads();          // publish stage kt across all waves

        const float* Ab = Alds[cur];
        const float* Bb = Blds[cur];
#pragma unroll
        for (int kk = 0; kk < 4; ++kk) {
            const int colf = kk * 4 + 2 * hh;  // even -> 8B aligned v2f loads
            v2f a[2], b[4];
#pragma unroll
            for (int sr = 0; sr < 2; ++sr)
                a[sr] = *(const v2f*)&Ab[(waveR * 32 + sr * 16 + nloc) * LSTR + colf];
#pragma unroll
            for (int sc = 0; sc < 4; ++sc)
                b[sc] = *(const v2f*)&Bb[(waveC * 64 + sc * 16 + nloc) * LSTR + colf];
#pragma unroll
            for (int sr = 0; sr < 2; ++sr)
#pragma unroll
                for (int sc = 0; sc < 4; ++sc)
                    acc[sr][sc] = __builtin_amdgcn_wmma_f32_16x16x4_f32(
                        false, a[sr], false, b[sc], (short)0, acc[sr][sc],
                        false, false);
        }
        __syncthreads();          // all waves done reading before buffer reuse
    }

    // ---- epilogue: dot -> distance -> exp, row/col reductions ----
    float sqj[4]; int tj4[4];
#pragma unroll
    for (int sc = 0; sc < 4; ++sc) {
        int j = jbase + waveC * 64 + sc * 16 + nloc;
        sqj[sc] = sqn[j];
        tj4[sc] = tgt[j];
    }
    float colacc[4] = {0.f, 0.f, 0.f, 0.f};

#pragma unroll
    for (int sr = 0; sr < 2; ++sr) {
#pragma unroll
        for (int p = 0; p < 8; ++p) {
            int irow = waveR * 32 + sr * 16 + p + 8 * hh;
            int i = ibase + irow;
            float sqi = sqn[i];
            int ti = tgt[i];
            float rsum = 0.0f;
#pragma unroll
            for (int sc = 0; sc < 4; ++sc) {
                float dot = acc[sr][sc][p];
                float sq = sqi + sqj[sc] - 2.0f * dot;
                sq = fmaxf(sq, 0.0f);
                float d = (sq > 1e-12f) ? sqrtf(sq) : 0.0f;
                float e = (ti != tj4[sc]) ? expf(MARGIN - d) : 0.0f;
                rsum += e;
                colacc[sc] += e;
            }
#pragma unroll
            for (int off = 8; off >= 1; off >>= 1)
                rsum += __shfl_xor(rsum, off, 16);
            if (nloc == 0) atomicAdd(&lnrow[irow], rsum);
        }
    }
#pragma unroll
    for (int sc = 0; sc < 4; ++sc) {
        float c = colacc[sc];
        c += __shfl_xor(c, 16, 32);
        if (hh == 0) atomicAdd(&lncol[waveC * 64 + sc * 16 + nloc], c);
    }

    __syncthreads();
    if (tid < TILE) {
        atomicAdd(&l_n[ibase + tid], lnrow[tid]);
        if (bi != bj) atomicAdd(&l_n[jbase + tid], lncol[tid]);
    }
}

// ----------------------------------------------------------- pair loss ------
__global__ __launch_bounds__(256)
void pair_loss_kernel(const float* __restrict__ X, const int* __restrict__ tgt,
                      float* __restrict__ ws) {
    const float* sqn = ws;
    const float* l_n = ws + N_EMB;
    float* accum = ws + 2 * N_EMB;

    int i = (int)((blockIdx.x * blockDim.x + threadIdx.x) >> 5);
    int lane = threadIdx.x & 31;
    if (i >= N_EMB) return;

    int ti = tgt[i];
    float lni = l_n[i];
    float sqi = sqn[i];

    float4 xi[4];
    const float4* Xi4 = (const float4*)(X + (size_t)i * D_EMB);
#pragma unroll
    for (int t = 0; t < 4; ++t) xi[t] = Xi4[lane + 32 * t];

    float lsum = 0.0f;
    int found = 0;
    for (int jb = i + 1; jb < N_EMB && found < NPOS; jb += 32) {
        int j = jb + lane;
        bool m = (j < N_EMB) && (tgt[j] == ti);
        unsigned mask = (unsigned)(__ballot(m) & 0xffffffffull);
        while (mask && found < NPOS) {
            int b = __ffs(mask) - 1;
            mask &= mask - 1;
            int jj = jb + b;              // wave-uniform
            const float4* Xj4 = (const float4*)(X + (size_t)jj * D_EMB);
            float dotp = 0.0f;
#pragma unroll
            for (int t = 0; t < 4; ++t) {
                float4 a = xi[t];
                float4 c = Xj4[lane + 32 * t];
                dotp += a.x * c.x + a.y * c.y + a.z * c.z + a.w * c.w;
            }
#pragma unroll
            for (int off = 16; off >= 1; off >>= 1)
                dotp += __shfl_xor(dotp, off, 32);
            float sq = sqi + sqn[jj] - 2.0f * dotp;
            sq = fmaxf(sq, 0.0f);
            float d = (sq > 1e-12f) ? sqrtf(sq) : 0.0f;
            float t2 = logf(lni + l_n[jj]) + d;
            t2 = fmaxf(t2, 0.0f);
            lsum += t2 * t2;
            ++found;
        }
    }
    if (lane == 0 && found) {
        atomicAdd(&accum[0], lsum);
        atomicAdd(&accum[1], (float)found);
    }
}

// ------------------------------------------------------------- finalize -----
__global__ void finalize_kernel(const float* __restrict__ ws,
                                float* __restrict__ out) {
    float loss = ws[2 * N_EMB + 0] / ws[2 * N_EMB + 1];
    float l2 = L2_REG * (ws[2 * N_EMB + 2] / (float)N_EMB);
    out[0] = loss + l2;
}

// ------------------------------------------------------------- launcher -----
extern "C" void kernel_launch(void* const* d_in, const int* in_sizes, int n_in,
                              void* d_out, int out_size, void* d_ws, size_t ws_size,
                              hipStream_t stream) {
    const float* X = (const float*)d_in[0];
    const int* tgt = (const int*)d_in[1];
    float* ws = (float*)d_ws;
    float* out = (float*)d_out;

    init_ws_kernel<<<(N_EMB + 255) / 256, 256, 0, stream>>>(ws);
    row_norms_kernel<<<(N_EMB * 32 + 255) / 256, 256, 0, stream>>>(X, ws);

    const int NB = N_EMB / TILE;                    // 32
    gram_exp_rowsum_kernel<<<NB * (NB + 1) / 2, 256, 0, stream>>>(X, tgt, ws);

    pair_loss_kernel<<<(N_EMB * 32 + 255) / 256, 256, 0, stream>>>(X, tgt, ws);
    finalize_kernel<<<1, 1, 0, stream>>>(ws, out);
}